// DilatedSpatioTemporalGCN_60129542620
// MI455X (gfx1250) — compile-verified
//
#include <hip/hip_runtime.h>
#include <hip/hip_bf16.h>

#define BSZ 4
#define TT  12
#define FD  32
#define NN  512
#define LL  3
#define KK  3
#define BT  (BSZ*TT)

typedef __attribute__((ext_vector_type(2))) float v2f;
typedef __attribute__((ext_vector_type(8))) float v8f;

// ---- workspace layout (float offsets) ----
#define OFF_E0 0
#define OFF_E1 786432
#define OFF_G  1572864
#define OFF_OA 2359296
#define OFF_XB 3145728   // xbuf: [b][n][f][l]  = 196608
#define OFF_SV 3342336   // sv:   [b][l*N+n]    = 6144
#define OFF_XS 3348480   // xsum: [bt][f]       = 1536
#define OFF_CV 3350016   // const:[bt][o]       = 1536
#define OFF_MM 3351552   // M:    [l][f][o]     = 3072
#define OFF_WK 3354624   // Wck:  [l][k][i][o]  = 9216

__device__ __forceinline__ v8f wmma_f32(v2f a, v2f b, v8f c) {
  return __builtin_amdgcn_wmma_f32_16x16x4_f32(false, a, false, b, (short)0, c,
                                               false, false);
}

// ---- precompute fused gate weight M = Ws@Wg_top + Wd@Wg_bot, and repack Wc ----
__global__ __launch_bounds__(256) void prep_kernel(
    const float* __restrict__ Wd, const float* __restrict__ Ws,
    const float* __restrict__ Wg, const float* __restrict__ Wc,
    float* __restrict__ MM, float* __restrict__ WK) {
  for (int idx = threadIdx.x; idx < LL * FD * FD; idx += blockDim.x) {
    int o = idx & 31, f = (idx >> 5) & 31, l = idx >> 10;
    float acc = 0.f;
    for (int f2 = 0; f2 < FD; ++f2) {
      acc += Ws[l * FD * FD + f * FD + f2] * Wg[f2 * FD + o];
      acc += Wd[l * FD * FD + f * FD + f2] * Wg[(FD + f2) * FD + o];
    }
    MM[idx] = acc;
  }
  for (int idx = threadIdx.x; idx < LL * KK * FD * FD; idx += blockDim.x) {
    int o = idx & 31, i = (idx >> 5) & 31, k = (idx >> 10) % KK,
        l = idx / (KK * FD * FD);
    WK[idx] = Wc[((l * FD + o) * FD + i) * KK + k];
  }
}

// ---- column sum of X over n (per bt,f): xsum[bt][f] = sum_n emb[bt][f][n] ----
__global__ __launch_bounds__(256) void xsum_kernel(const float* __restrict__ emb,
                                                   float* __restrict__ xs) {
  int w = (blockIdx.x * 256 + threadIdx.x) >> 5;  // 1536 waves
  int lane = threadIdx.x & 31;
  int bt = w >> 5, f = w & 31;
  const float* p = emb + (size_t)bt * FD * NN + f * NN;
  float s = 0.f;
  for (int j = lane; j < NN; j += 32) s += p[j];
  for (int off = 16; off; off >>= 1) s += __shfl_down(s, off, 32);
  if (lane == 0) xs[bt * FD + f] = s;
}

// ---- per-(b,t) gate constant vector ----
__global__ void const_kernel(const float* __restrict__ xs,
                             const float* __restrict__ MM,
                             const float* __restrict__ Wg,
                             const float* __restrict__ bs,
                             const float* __restrict__ bd,
                             const float* __restrict__ bg,
                             float* __restrict__ cv, int l) {
  int bt = blockIdx.x, o = threadIdx.x;
  const float inv513 = 1.0f / 513.0f;
  const float* M = MM + l * FD * FD;
  float c = bg[o];
  for (int f = 0; f < FD; ++f) {
    c += xs[bt * FD + f] * inv513 * M[f * FD + o];
    c += bs[l * FD + f] * Wg[f * FD + o];
    c += bd[l * FD + f] * Wg[(FD + f) * FD + o];
  }
  cv[bt * FD + o] = c;
}

// ---- gate GEMM: g[bt][n][o] = sigmoid((X@M)/513 + const) via WMMA f32 16x16x4 ----
__global__ __launch_bounds__(128) void gate_kernel(
    const float* __restrict__ emb, const float* __restrict__ MM,
    const float* __restrict__ cv, float* __restrict__ g, int l) {
  int wid = threadIdx.x >> 5, lane = threadIdx.x & 31;
  int bt = blockIdx.x >> 3, nblk = blockIdx.x & 7;
  int n0 = nblk * 64 + wid * 16;
  int m = lane & 15, hi = lane >> 4;
  const float* A = emb + (size_t)bt * FD * NN;  // A[n][f] = emb[bt][f][n]
  const float* B = MM + l * FD * FD;            // B[f][o]
  v8f acc0 = {}, acc1 = {};
  for (int k0 = 0; k0 < FD; k0 += 4) {
    int kk = k0 + hi * 2;
    v2f a;  a.x = A[kk * NN + n0 + m];        a.y = A[(kk + 1) * NN + n0 + m];
    v2f b0; b0.x = B[kk * FD + m];            b0.y = B[(kk + 1) * FD + m];
    v2f b1; b1.x = B[kk * FD + 16 + m];       b1.y = B[(kk + 1) * FD + 16 + m];
    acc0 = wmma_f32(a, b0, acc0);
    acc1 = wmma_f32(a, b1, acc1);
  }
  const float inv513 = 1.0f / 513.0f;
  float c0 = cv[bt * FD + m], c1 = cv[bt * FD + 16 + m];
  float* go = g + (size_t)bt * NN * FD;
#pragma unroll
  for (int r = 0; r < 8; ++r) {
    int row = n0 + r + hi * 8;
    float x0 = acc0[r] * inv513 + c0;
    float x1 = acc1[r] * inv513 + c1;
    go[row * FD + m]      = 1.0f / (1.0f + __expf(-x0));
    go[row * FD + 16 + m] = 1.0f / (1.0f + __expf(-x1));
  }
}

// ---- causal dilated conv as 3 accumulated GEMMs: oa[bt][n][o] (pre-activation) ----
__global__ __launch_bounds__(128) void conv_kernel(const float* __restrict__ g,
                                                   const float* __restrict__ WK,
                                                   float* __restrict__ oa, int l) {
  int wid = threadIdx.x >> 5, lane = threadIdx.x & 31;
  int bt = blockIdx.x >> 3, nblk = blockIdx.x & 7;
  int n0 = nblk * 64 + wid * 16;
  int m = lane & 15, hi = lane >> 4;
  int t = bt % TT, b = bt / TT;
  int dil = 1 << l;
  v8f acc0 = {}, acc1 = {};
  for (int k = 0; k < KK; ++k) {
    int ts = t - (KK - 1 - k) * dil;
    if (ts < 0) continue;  // wave-uniform
    const float* A = g + (size_t)(b * TT + ts) * NN * FD;  // A[n][i]
    const float* B = WK + (l * KK + k) * FD * FD;          // B[i][o]
    for (int k0 = 0; k0 < FD; k0 += 4) {
      int kk = k0 + hi * 2;
      v2f a;  a.x = A[(n0 + m) * FD + kk];  a.y = A[(n0 + m) * FD + kk + 1];
      v2f b0; b0.x = B[kk * FD + m];        b0.y = B[(kk + 1) * FD + m];
      v2f b1; b1.x = B[kk * FD + 16 + m];   b1.y = B[(kk + 1) * FD + 16 + m];
      acc0 = wmma_f32(a, b0, acc0);
      acc1 = wmma_f32(a, b1, acc1);
    }
  }
  float* out = oa + (size_t)bt * NN * FD;
#pragma unroll
  for (int r = 0; r < 8; ++r) {
    int row = n0 + r + hi * 8;
    out[row * FD + m]      = acc0[r];
    out[row * FD + 16 + m] = acc1[r];
  }
}

// ---- bias + relu + residual (transposed store) + res_agg capture at t=T-1 ----
__global__ __launch_bounds__(256) void finish_kernel(
    const float* __restrict__ oa, const float* __restrict__ embin,
    const float* __restrict__ bc, float* __restrict__ embout,
    float* __restrict__ xbuf, int l) {
  int idx = blockIdx.x * 256 + threadIdx.x;  // BT*FD*N
  int n = idx & (NN - 1);
  int f = (idx >> 9) & (FD - 1);
  int bt = idx >> 14;
  float val = oa[(size_t)bt * NN * FD + n * FD + f] + bc[l * FD + f];
  val = fmaxf(val, 0.0f);
  size_t ei = (size_t)bt * FD * NN + (size_t)f * NN + n;
  embout[ei] = val + embin[ei];
  int t = bt % TT, b = bt / TT;
  if (t == TT - 1) xbuf[(((size_t)b * NN + n) * FD + f) * LL + l] = val;
}

// ---- sv[b][l*N+n] = sum_d tanh(x[b,n,:,l]@Wa + ba)[d] * v[d] ----
__global__ __launch_bounds__(256) void sv_kernel(const float* __restrict__ xbuf,
                                                 const float* __restrict__ Wa,
                                                 const float* __restrict__ ba,
                                                 const float* __restrict__ v,
                                                 float* __restrict__ sv) {
  int tid = blockIdx.x * 256 + threadIdx.x;  // BSZ*L*N = 6144
  int n = tid & (NN - 1);
  int li = (tid >> 9) % LL;
  int b = tid / (LL * NN);
  float xr[FD];
#pragma unroll
  for (int kk = 0; kk < FD; ++kk)
    xr[kk] = xbuf[(((size_t)b * NN + n) * FD + kk) * LL + li];
  float acc = 0.f;
  for (int d = 0; d < FD; ++d) {
    float s = ba[d];
#pragma unroll
    for (int kk = 0; kk < FD; ++kk) s += xr[kk] * Wa[kk * FD + d];
    acc += tanhf(s) * v[d];
  }
  sv[(size_t)b * LL * NN + li * NN + n] = acc;
}

// ---- scrambled-reshape softmax over flat triples + weighted combine ----
__global__ __launch_bounds__(256) void out_kernel(const float* __restrict__ sv,
                                                  const float* __restrict__ xbuf,
                                                  float* __restrict__ Y) {
  int tid = blockIdx.x * 256 + threadIdx.x;  // BSZ*N = 2048
  int n = tid & (NN - 1);
  int b = tid >> 9;
  const float* s = sv + (size_t)b * LL * NN + 3 * n;  // sr = flat reshape
  float e0 = s[0], e1 = s[1], e2 = s[2];
  float mx = fmaxf(e0, fmaxf(e1, e2));
  float a0 = __expf(e0 - mx), a1 = __expf(e1 - mx), a2 = __expf(e2 - mx);
  float inv = 1.0f / (a0 + a1 + a2);
  a0 *= inv; a1 *= inv; a2 *= inv;
  const float* xb = xbuf + ((size_t)b * NN + n) * FD * LL;
  float* y = Y + ((size_t)b * NN + n) * FD;
#pragma unroll
  for (int d = 0; d < FD; ++d)
    y[d] = a0 * xb[d * LL] + a1 * xb[d * LL + 1] + a2 * xb[d * LL + 2];
}

extern "C" void kernel_launch(void* const* d_in, const int* in_sizes, int n_in,
                              void* d_out, int out_size, void* d_ws, size_t ws_size,
                              hipStream_t stream) {
  (void)in_sizes; (void)n_in; (void)out_size; (void)ws_size;
  const float* emb = (const float*)d_in[0];
  // d_in[1] (B), d_in[2] (static_MTE), d_in[14] (use_MTE) are provably dead:
  // softmax outputs are strictly positive, so gcn_norm yields the constant (J+I)/513.
  const float* Wd = (const float*)d_in[3];
  const float* bd = (const float*)d_in[4];
  const float* Ws = (const float*)d_in[5];
  const float* bs = (const float*)d_in[6];
  const float* Wg = (const float*)d_in[7];
  const float* bg = (const float*)d_in[8];
  const float* Wc = (const float*)d_in[9];
  const float* bc = (const float*)d_in[10];
  const float* Wa = (const float*)d_in[11];
  const float* ba = (const float*)d_in[12];
  const float* vv = (const float*)d_in[13];

  float* ws = (float*)d_ws;
  float* E0 = ws + OFF_E0;
  float* E1 = ws + OFF_E1;
  float* G  = ws + OFF_G;
  float* OA = ws + OFF_OA;
  float* XB = ws + OFF_XB;
  float* SV = ws + OFF_SV;
  float* XS = ws + OFF_XS;
  float* CV = ws + OFF_CV;
  float* MM = ws + OFF_MM;
  float* WK = ws + OFF_WK;

  prep_kernel<<<1, 256, 0, stream>>>(Wd, Ws, Wg, Wc, MM, WK);

  const float* cur = emb;
  float* bufs[2] = {E0, E1};
  for (int l = 0; l < LL; ++l) {
    xsum_kernel<<<192, 256, 0, stream>>>(cur, XS);
    const_kernel<<<BT, 32, 0, stream>>>(XS, MM, Wg, bs, bd, bg, CV, l);
    gate_kernel<<<BT * 8, 128, 0, stream>>>(cur, MM, CV, G, l);
    conv_kernel<<<BT * 8, 128, 0, stream>>>(G, WK, OA, l);
    float* nxt = bufs[l & 1];
    finish_kernel<<<(BT * FD * NN) / 256, 256, 0, stream>>>(OA, cur, bc, nxt, XB, l);
    cur = nxt;
  }

  sv_kernel<<<(BSZ * LL * NN) / 256, 256, 0, stream>>>(XB, Wa, ba, vv, SV);
  out_kernel<<<(BSZ * NN) / 256, 256, 0, stream>>>(SV, XB, (float*)d_out);
}